// LM_MLE_48241072669067
// MI455X (gfx1250) — compile-verified
//
#include <hip/hip_runtime.h>
#include <math.h>

typedef __attribute__((ext_vector_type(2))) float v2f;
typedef __attribute__((ext_vector_type(8))) float v8f;

#define NBATCH 4096
#define SDIM   1024
#define KDIM   8
#define TPB    128   // 4 waves of 32
#define BPB    16    // batches per block
#define NITER  3
#define LAMBDA 0.01f

__global__ __launch_bounds__(TPB) void lm_mle_wmma_kernel(
    const float* __restrict__ smp,      // [4096,1024]
    const float* __restrict__ initial,  // [4096,8]
    const float* __restrict__ basis,    // [1024,8]
    const float* __restrict__ prm,      // [8,2] min/max
    float* __restrict__ out)            // [4096,8]
{
  __shared__ __align__(16) float sBasis[SDIM * KDIM];   // 32 KB
  __shared__ float sPart[4 * BPB * KDIM * KDIM];        // 16 KB (per-wave alpha partials)
  __shared__ float sAlpha[BPB * KDIM * KDIM];           // 4 KB
  __shared__ float sColsq[BPB * KDIM];
  __shared__ float sDiagAdd[BPB * KDIM];
  __shared__ float sCur[BPB * KDIM];
  __shared__ float sBeta[BPB * KDIM];
  __shared__ float sBPart[4 * BPB * KDIM];              // per-wave beta partials

  const int tid   = threadIdx.x;
  const int lane  = tid & 31;
  const int wave  = tid >> 5;
  const int bBase = blockIdx.x * BPB;
  const int n     = lane & 15;   // column / batch-row index within tile
  const int h     = lane >> 4;   // half selector: k = v + 2*h

  // ---- stage basis into LDS (row-major [s][k], 32B rows for b128 loads)
  for (int i = tid; i < SDIM * KDIM; i += TPB) sBasis[i] = basis[i];
  // ---- init cur = initial
  if (tid < BPB * KDIM) sCur[tid] = initial[bBase * KDIM + tid];
  __syncthreads();

  const int sBeg = wave * (SDIM / 4);   // each wave owns 256 s-values

  // ================= precompute alpha[b] = basis^T diag(smp[b]) basis =====
  // 8 WMMA chains, 2 batches each. A = basis^T (rows 0-7), B = smp .* basis.
  v8f accA[8];
#pragma unroll
  for (int p = 0; p < 8; ++p) accA[p] = v8f{0, 0, 0, 0, 0, 0, 0, 0};

  for (int c = 0; c < 64; ++c) {
    const int s0 = sBeg + 4 * c;
    const int sa = s0 + 2 * h;                 // s' for v=0 (v=1 is sa+1)
    const float be0 = sBasis[sa * KDIM + (n & 7)];
    const float be1 = sBasis[(sa + 1) * KDIM + (n & 7)];
    v2f a;
    a[0] = (n < 8) ? be0 : 0.0f;               // A[m,k] = basis[s', m], m<8
    a[1] = (n < 8) ? be1 : 0.0f;
#pragma unroll
    for (int p = 0; p < 8; ++p) {
      const float2 sp = *reinterpret_cast<const float2*>(
          smp + (bBase + 2 * p + (n >> 3)) * SDIM + sa);
      v2f bv;
      bv[0] = sp.x * be0;                      // B[k,n] = smp[b_{n/8},s'] * basis[s', n%8]
      bv[1] = sp.y * be1;
      accA[p] = __builtin_amdgcn_wmma_f32_16x16x4_f32(
          false, a, false, bv, (short)0, accA[p], false, false);
    }
  }
  // D[i<8, j]: alpha[b_{j/8}][i][j%8]; valid rows live in lanes 0-15 (m = r)
  if (lane < 16) {
#pragma unroll
    for (int p = 0; p < 8; ++p)
#pragma unroll
      for (int r = 0; r < 8; ++r)
        sPart[wave * (BPB * 64) + (2 * p + (n >> 3)) * 64 + r * 8 + (n & 7)] =
            accA[p][r];
  }
  __syncthreads();
  // cross-wave reduce into sAlpha (1024 values, 8 per thread)
#pragma unroll
  for (int j = 0; j < 8; ++j) {
    const int o = tid * 8 + j;
    sAlpha[o] = sPart[o] + sPart[1024 + o] + sPart[2048 + o] + sPart[3072 + o];
  }
  __syncthreads();

  // ---- damping diagonal (constant across iterations):
  // scale[b,j] = sum_i alpha[b,i,j]^2 ; scale /= mean_j ; diag += LAMBDA*scale
  {
    const int b = tid >> 3, j = tid & 7;
    float cs = 0.0f;
#pragma unroll
    for (int i = 0; i < KDIM; ++i) {
      const float v = sAlpha[b * 64 + i * 8 + j];
      cs += v * v;
    }
    sColsq[tid] = cs;
  }
  __syncthreads();
  {
    float m = 0.0f;
#pragma unroll
    for (int k = 0; k < KDIM; ++k) m += sColsq[(tid & ~7) + k];
    m *= 0.125f;
    sDiagAdd[tid] = LAMBDA * sColsq[tid] / m;
  }
  __syncthreads();

  // ========================= LM iterations ===============================
  for (int it = 0; it < NITER; ++it) {
    // per-lane copy of cur for batch m = n (A-matrix row index)
    float curk[KDIM];
#pragma unroll
    for (int k = 0; k < KDIM; ++k) curk[k] = sCur[n * KDIM + k];

    v8f accB = v8f{0, 0, 0, 0, 0, 0, 0, 0};
    for (int c = 0; c < 64; ++c) {
      const int s0 = sBeg + 4 * c;
      const int sa = s0 + 2 * h;
      const float4* r0 = reinterpret_cast<const float4*>(&sBasis[sa * KDIM]);
      const float4* r1 = reinterpret_cast<const float4*>(&sBasis[(sa + 1) * KDIM]);
      const float4 r0a = r0[0], r0b = r0[1];
      const float4 r1a = r1[0], r1b = r1[1];
      const float eta0 = curk[0] * r0a.x + curk[1] * r0a.y + curk[2] * r0a.z +
                         curk[3] * r0a.w + curk[4] * r0b.x + curk[5] * r0b.y +
                         curk[6] * r0b.z + curk[7] * r0b.w;
      const float eta1 = curk[0] * r1a.x + curk[1] * r1a.y + curk[2] * r1a.z +
                         curk[3] * r1a.w + curk[4] * r1b.x + curk[5] * r1b.y +
                         curk[6] * r1b.z + curk[7] * r1b.w;
      const float2 sp = *reinterpret_cast<const float2*>(
          smp + (bBase + n) * SDIM + sa);
      v2f a;
      a[0] = sp.x - expf(eta0);                // A[m,k] = smp - mu  (batch m = n)
      a[1] = sp.y - expf(eta1);
      v2f bv;
      bv[0] = sBasis[sa * KDIM + (n & 7)];     // B[k,n] = basis[s', n%8]
      bv[1] = sBasis[(sa + 1) * KDIM + (n & 7)];
      accB = __builtin_amdgcn_wmma_f32_16x16x4_f32(
          false, a, false, bv, (short)0, accB, false, false);
    }
    // D[m,n<8] = beta[b=m][n]; m = r + 8*h
    if ((lane & 15) < 8) {
#pragma unroll
      for (int r = 0; r < 8; ++r)
        sBPart[wave * (BPB * KDIM) + (r + 8 * h) * KDIM + (lane & 7)] = accB[r];
    }
    __syncthreads();
    if (tid < BPB * KDIM)
      sBeta[tid] = sBPart[tid] + sBPart[128 + tid] + sBPart[256 + tid] +
                   sBPart[384 + tid];
    __syncthreads();

    // ---- per-batch damped 8x8 solve + clamped update (lanes 0-15)
    if (tid < BPB) {
      float M[KDIM][KDIM], rhs[KDIM], x[KDIM];
#pragma unroll
      for (int i = 0; i < KDIM; ++i) {
#pragma unroll
        for (int j = 0; j < KDIM; ++j) M[i][j] = sAlpha[tid * 64 + i * 8 + j];
        M[i][i] += sDiagAdd[tid * 8 + i];
        rhs[i] = sBeta[tid * 8 + i];
      }
#pragma unroll
      for (int p = 0; p < KDIM; ++p) {
        const float inv = 1.0f / M[p][p];
#pragma unroll
        for (int r = p + 1; r < KDIM; ++r) {
          const float f = M[r][p] * inv;
#pragma unroll
          for (int cc = p; cc < KDIM; ++cc) M[r][cc] -= f * M[p][cc];
          rhs[r] -= f * rhs[p];
        }
      }
#pragma unroll
      for (int p = KDIM - 1; p >= 0; --p) {
        float acc = rhs[p];
#pragma unroll
        for (int cc = p + 1; cc < KDIM; ++cc) acc -= M[p][cc] * x[cc];
        x[p] = acc / M[p][p];
      }
#pragma unroll
      for (int k = 0; k < KDIM; ++k) {
        float v = sCur[tid * 8 + k] + x[k];
        v = fmaxf(v, prm[2 * k + 0]);
        v = fminf(v, prm[2 * k + 1]);
        sCur[tid * 8 + k] = v;
      }
    }
    __syncthreads();
  }

  if (tid < BPB * KDIM) out[bBase * KDIM + tid] = sCur[tid];
}

extern "C" void kernel_launch(void* const* d_in, const int* in_sizes, int n_in,
                              void* d_out, int out_size, void* d_ws,
                              size_t ws_size, hipStream_t stream) {
  (void)in_sizes; (void)n_in; (void)out_size; (void)d_ws; (void)ws_size;
  const float* smp     = (const float*)d_in[0];
  const float* initial = (const float*)d_in[1];
  const float* basis   = (const float*)d_in[2];
  const float* prm     = (const float*)d_in[3];
  float* out = (float*)d_out;
  lm_mle_wmma_kernel<<<dim3(NBATCH / BPB), dim3(TPB), 0, stream>>>(
      smp, initial, basis, prm, out);
}